// EdgeEncoding_57655640982216
// MI455X (gfx1250) — compile-verified
//
#include <hip/hip_runtime.h>
#include <float.h>

// ---------------------------------------------------------------------------
// EdgeEncoding tensor path: out = nan_to_num(min(weights, 5) * mean(edge_vector))
//
// weights: [8192, 8192] f32 (256 MB)  ->  out: same shape (256 MB)
// Pure streaming elementwise op: 512 MB of single-touch traffic, ~22 us floor
// at 23.3 TB/s HBM. No matmul structure -> no WMMA (scalar broadcast multiply).
// Strategy: B128 non-temporal loads/stores (bypass 192MB L2 for 512MB stream),
// 4 outstanding loads per thread, massive wave32 occupancy.
// ---------------------------------------------------------------------------

typedef __attribute__((ext_vector_type(4))) float v4f;

#define MAX_PATH_DISTANCE 5.0f

__device__ __forceinline__ float edge_fix1(float w, float m) {
    // clamp-above at 5, scale by mean, then nan_to_num (NaN->0, inf->+/-FLT_MAX)
    float y = (w < MAX_PATH_DISTANCE ? w : MAX_PATH_DISTANCE) * m;
    y = (y != y) ? 0.0f : y;          // NaN -> 0
    y = (y >  FLT_MAX) ?  FLT_MAX : y; // +inf -> max float
    y = (y < -FLT_MAX) ? -FLT_MAX : y; // -inf -> lowest float
    return y;
}

// Tiny single-block reduction: mean of edge_vector (128 elems) into d_ws[0].
// wave32 shuffle reduction + LDS combine across waves.
__global__ void ev_mean_kernel(const float* __restrict__ v, int n,
                               float* __restrict__ out) {
    __shared__ float red[32];
    float s = 0.0f;
    for (int i = threadIdx.x; i < n; i += blockDim.x) s += v[i];
#pragma unroll
    for (int o = 16; o > 0; o >>= 1) s += __shfl_down(s, o, 32);
    const int lane = threadIdx.x & 31;
    const int wid  = threadIdx.x >> 5;
    if (lane == 0) red[wid] = s;
    __syncthreads();
    if (threadIdx.x == 0) {
        float t = 0.0f;
        const int nw = ((int)blockDim.x + 31) >> 5;
        for (int i = 0; i < nw; ++i) t += red[i];
        *out = t / (float)n;
    }
}

// Streaming elementwise kernel: grid-stride over float4, unrolled x4 so each
// thread keeps 4 B128 NT loads in flight.
__global__ void edge_encoding_kernel(const float* __restrict__ w,
                                     const float* __restrict__ mean_p,
                                     float* __restrict__ out,
                                     long long n) {
    const float m = *mean_p;  // uniform; hot in cache after first wave
    const long long n4     = n >> 2;
    const long long stride = (long long)gridDim.x * blockDim.x;
    long long i = (long long)blockIdx.x * blockDim.x + threadIdx.x;

    const v4f* __restrict__ wv = (const v4f*)w;
    v4f* __restrict__       ov = (v4f*)out;

    // Main loop: 4 outstanding 128-bit non-temporal loads per thread.
    for (; i + 3 * stride < n4; i += 4 * stride) {
        v4f a0 = __builtin_nontemporal_load(&wv[i]);
        v4f a1 = __builtin_nontemporal_load(&wv[i +     stride]);
        v4f a2 = __builtin_nontemporal_load(&wv[i + 2 * stride]);
        v4f a3 = __builtin_nontemporal_load(&wv[i + 3 * stride]);
#pragma unroll
        for (int j = 0; j < 4; ++j) {
            a0[j] = edge_fix1(a0[j], m);
            a1[j] = edge_fix1(a1[j], m);
            a2[j] = edge_fix1(a2[j], m);
            a3[j] = edge_fix1(a3[j], m);
        }
        __builtin_nontemporal_store(a0, &ov[i]);
        __builtin_nontemporal_store(a1, &ov[i +     stride]);
        __builtin_nontemporal_store(a2, &ov[i + 2 * stride]);
        __builtin_nontemporal_store(a3, &ov[i + 3 * stride]);
    }
    // Remainder float4s
    for (; i < n4; i += stride) {
        v4f a = __builtin_nontemporal_load(&wv[i]);
#pragma unroll
        for (int j = 0; j < 4; ++j) a[j] = edge_fix1(a[j], m);
        __builtin_nontemporal_store(a, &ov[i]);
    }
    // Scalar tail (n % 4), handled by the first few threads of the grid.
    const long long base = n4 << 2;
    const long long gid  = (long long)blockIdx.x * blockDim.x + threadIdx.x;
    if (base + gid < n) out[base + gid] = edge_fix1(w[base + gid], m);
}

extern "C" void kernel_launch(void* const* d_in, const int* in_sizes, int n_in,
                              void* d_out, int out_size, void* d_ws, size_t ws_size,
                              hipStream_t stream) {
    // setup_inputs order: 0=x, 1=edge_attr, 2=weights, 3=edge_vector
    const float* weights     = (const float*)d_in[2];
    const float* edge_vector = (const float*)d_in[3];
    const int    ev_n        = in_sizes[3];

    float* mean_ws = (float*)d_ws;  // one scalar of scratch
    float* out     = (float*)d_out;
    const long long n = (long long)out_size;  // 8192*8192

    // 1) scalar mean of edge_vector
    ev_mean_kernel<<<1, 128, 0, stream>>>(edge_vector, ev_n, mean_ws);

    // 2) streaming clamp-scale-fixup over 64M floats
    const int threads = 256;  // 8 wave32s per block
    long long n4 = n >> 2;
    long long want = (n4 + (long long)(threads * 8) - 1) / (long long)(threads * 8);
    if (want < 1) want = 1;
    const int blocks = (int)(want < 16384 ? want : 16384);

    edge_encoding_kernel<<<blocks, threads, 0, stream>>>(weights, mean_ws, out, n);
}